// ChamferDistanceL2_81475529605150
// MI455X (gfx1250) — compile-verified
//
#include <hip/hip_runtime.h>
#include <hip/hip_bf16.h>

// ---------------------------------------------------------------------------
// Chamfer L2 (B=4, N=M=8192, 3-D points) for gfx1250.
//
// Cross term is a K=3 GEMM -> V_WMMA_F32_16X16X4_F32 with the K=4 pad slot
// used to fold ||y||^2 in:  A row i = [-2x0,-2x1,-2x2, 1], B col j =
// [y0,y1,y2,||y_j||^2], C = 0  =>  D = ||y_j||^2 - 2 x_i.y_j  directly.
// Running min kept across the M loop with v_min3_num_f32 (two column tiles
// per iteration, no NaN-canonicalize overhead); ||x_i||^2 + clamp applied
// once at the end (max(.,0) commutes with min). Deterministic: per-block
// partial sums to fixed workspace slots, fixed-order final reduce.
// ---------------------------------------------------------------------------

typedef float v2f __attribute__((ext_vector_type(2)));
typedef float v8f __attribute__((ext_vector_type(8)));

namespace {
constexpr int FRAGS           = 2;                        // 16-row A fragments per wave
constexpr int WAVES_PER_BLOCK = 8;                        // 256 threads (wave32)
constexpr int ROWS_PER_WAVE   = FRAGS * 16;               // 32
constexpr int ROWS_PER_BLOCK  = WAVES_PER_BLOCK * ROWS_PER_WAVE; // 256
}

// One launch handles BOTH chamfer directions (gridDim.z = 2) for occupancy:
//   dir 0: X = xyz1 (rows), Y = xyz2 (cols)  -> dist1 row-mins
//   dir 1: X = xyz2 (rows), Y = xyz1 (cols)  -> dist2 row-mins
// Fused grid requires N == M (true here: 8192 == 8192).
__global__ void __launch_bounds__(WAVES_PER_BLOCK * 32, 1)
chamfer_min_rows_kernel(const float* __restrict__ xyz1,
                        const float* __restrict__ xyz2,
                        int N, int M,
                        float* __restrict__ partials, int nbX)
{
    const int dir = blockIdx.z;
    const int b   = blockIdx.y;
    const int nB  = gridDim.y;

    const float* __restrict__ X = dir ? (xyz2 + (size_t)b * M * 3)
                                      : (xyz1 + (size_t)b * N * 3);
    const float* __restrict__ Y = dir ? (xyz1 + (size_t)b * N * 3)
                                      : (xyz2 + (size_t)b * M * 3);
    const int nY = dir ? N : M;

    __shared__ float sq1s[ROWS_PER_BLOCK];     // ||x_i||^2 for this block's rows
    __shared__ float wavesum[WAVES_PER_BLOCK];

    const int tid      = threadIdx.x;
    const int blockRow = blockIdx.x * ROWS_PER_BLOCK;

    // Precompute ||x||^2 for the block's 256 rows.
    for (int i = tid; i < ROWS_PER_BLOCK; i += blockDim.x) {
        const float* xp = X + (size_t)(blockRow + i) * 3;
        float x0 = xp[0], x1 = xp[1], x2 = xp[2];
        sq1s[i] = x0 * x0 + x1 * x1 + x2 * x2;
    }
    __syncthreads();

    const int wave    = tid >> 5;
    const int lane    = tid & 31;
    const int half    = lane >> 4;    // K-pair select for A/B fragments
    const int lcol    = lane & 15;    // row-in-fragment (A) / column (B,C,D)
    const int waveRow = blockRow + wave * ROWS_PER_WAVE;

    // A fragments: 16x4 fp32, row = lcol.  Layout (wave32):
    //   lanes 0-15 : VGPR0=K0, VGPR1=K1 ; lanes 16-31 : VGPR0=K2, VGPR1=K3.
    // K0..K2 = -2*x, K3 = 1 (multiplies ||y||^2 carried in B's K3 row).
    v2f a[FRAGS];
#pragma unroll
    for (int f = 0; f < FRAGS; ++f) {
        const float* xp = X + (size_t)(waveRow + f * 16 + lcol) * 3;
        if (half == 0) { a[f][0] = -2.0f * xp[0]; a[f][1] = -2.0f * xp[1]; }
        else           { a[f][0] = -2.0f * xp[2]; a[f][1] = 1.0f; }
    }

    const v8f czero = {};              // loop-invariant C = 0

    // Running per-(row, col%16) minima, accumulator layout, plain scalars so
    // they can be inline-asm min3 destinations.
    float acc[FRAGS][8];
#pragma unroll
    for (int f = 0; f < FRAGS; ++f)
#pragma unroll
        for (int v = 0; v < 8; ++v) acc[f][v] = 3.0e38f;

    // Two 16-column tiles per iteration (32 columns).
    for (int j0 = 0; j0 < nY; j0 += 32) {
        const float* yp = Y + (size_t)(j0 + lcol) * 3;
        float y0a = yp[0],  y1a = yp[1],  y2a = yp[2];   // tile 0: cols j0+lcol
        float y0b = yp[48], y1b = yp[49], y2b = yp[50];  // tile 1: +16 points

        __builtin_prefetch(yp + 96, 0, 3);               // next 32 points

        const float s2a = y0a * y0a + y1a * y1a + y2a * y2a;
        const float s2b = y0b * y0b + y1b * y1b + y2b * y2b;

        // B fragments: 4x16 fp32, column = lcol.
        //   lanes 0-15 : VGPR0=K0, VGPR1=K1 ; lanes 16-31 : VGPR0=K2, VGPR1=K3.
        // K3 row carries ||y_col||^2.
        v2f bfa, bfb;
        if (half == 0) { bfa[0] = y0a; bfa[1] = y1a; bfb[0] = y0b; bfb[1] = y1b; }
        else           { bfa[0] = y2a; bfa[1] = s2a; bfb[0] = y2b; bfb[1] = s2b; }

#pragma unroll
        for (int f = 0; f < FRAGS; ++f) {
            v8f d0 = __builtin_amdgcn_wmma_f32_16x16x4_f32(
                false, a[f], false, bfa, (short)0, czero, false, false);
            v8f d1 = __builtin_amdgcn_wmma_f32_16x16x4_f32(
                false, a[f], false, bfb, (short)0, czero, false, false);
            // acc = min(acc, d0, d1) without LLVM's NaN-canonicalize traffic.
#pragma unroll
            for (int v = 0; v < 8; ++v)
                asm("v_min3_num_f32 %0, %0, %1, %2"
                    : "+v"(acc[f][v])
                    : "v"(d0[v]), "v"(d1[v]));
        }
    }

    // D layout (16x16 f32): VGPR v, lanes 0-15 -> row v, lanes 16-31 -> row v+8,
    // column = lane%16. Min across the 16 columns via shfl-xor in each half,
    // then add ||x||^2, clamp at 0, and sum the block's row-mins.
    float lanesum = 0.0f;
#pragma unroll
    for (int f = 0; f < FRAGS; ++f) {
#pragma unroll
        for (int v = 0; v < 8; ++v) {
            float m = acc[f][v];
            m = fminf(m, __shfl_xor(m, 1, 16));
            m = fminf(m, __shfl_xor(m, 2, 16));
            m = fminf(m, __shfl_xor(m, 4, 16));
            m = fminf(m, __shfl_xor(m, 8, 16));
            const int localRow = wave * ROWS_PER_WAVE + f * 16 + v + 8 * half;
            const float rowmin = fmaxf(m + sq1s[localRow], 0.0f);
            if (lcol == 0) lanesum += rowmin;   // one lane per half contributes
        }
    }
    lanesum += __shfl_xor(lanesum, 16, 32);     // combine the two halves
    if (lane == 0) wavesum[wave] = lanesum;
    __syncthreads();

    if (tid == 0) {
        float s = 0.0f;
#pragma unroll
        for (int w = 0; w < WAVES_PER_BLOCK; ++w) s += wavesum[w];
        partials[((size_t)dir * nB + b) * nbX + blockIdx.x] = s;
    }
}

// Fixed-order scalar reduce: out = sum(dir0)/(B*N) + sum(dir1)/(B*M).
__global__ void chamfer_finalize_kernel(const float* __restrict__ partials,
                                        int nPerDir, float inv1, float inv2,
                                        float* __restrict__ out)
{
    if (threadIdx.x == 0 && blockIdx.x == 0) {
        float s1 = 0.0f, s2 = 0.0f;
        for (int i = 0; i < nPerDir; ++i) s1 += partials[i];
        for (int i = 0; i < nPerDir; ++i) s2 += partials[nPerDir + i];
        out[0] = s1 * inv1 + s2 * inv2;
    }
}

extern "C" void kernel_launch(void* const* d_in, const int* in_sizes, int n_in,
                              void* d_out, int out_size, void* d_ws, size_t ws_size,
                              hipStream_t stream) {
    (void)n_in; (void)out_size; (void)ws_size;
    const float* xyz1 = (const float*)d_in[0];   // (B, N, 3) fp32
    const float* xyz2 = (const float*)d_in[1];   // (B, M, 3) fp32

    const int B = 4;
    const int N = in_sizes[0] / (B * 3);         // 8192
    const int M = in_sizes[1] / (B * 3);         // 8192

    const int nbX     = N / ROWS_PER_BLOCK;      // 32 row-blocks per batch
    const int nPerDir = B * nbX;                 // 128 partials per direction

    float* partials = (float*)d_ws;              // 2 * 128 floats of scratch

    dim3 grid(nbX, B, 2), block(WAVES_PER_BLOCK * 32);
    chamfer_min_rows_kernel<<<grid, block, 0, stream>>>(
        xyz1, xyz2, N, M, partials, nbX);

    chamfer_finalize_kernel<<<1, 32, 0, stream>>>(
        partials, nPerDir,
        1.0f / ((float)B * (float)N),
        1.0f / ((float)B * (float)M),
        (float*)d_out);
}